// MaxPoolAggregation_26087631356321
// MI455X (gfx1250) — compile-verified
//
#include <hip/hip_runtime.h>
#include <stdint.h>

// Problem constants (from reference): N=8192 nodes, D=128 features, f32.
#define N_NODES 8192
#define D_FEAT  128
#define CCOLS   32                          // adjacency columns per chunk
#define NCHUNK  (N_NODES / CCOLS)           // 256
#define GROUPS  (CCOLS / 4)                 // 8 groups of 4 columns per chunk
#define BLOCK   256                         // 8 waves (wave32)
#define ROWS_PER_WAVE 4
#define ROWS_PER_WG   32                    // 8 waves * 4 rows
#define GRID    (N_NODES / ROWS_PER_WG)     // 256 workgroups

// Dynamic LDS layout (in floats):
//   T0: [8 groups][16 masks][128 feats] = 16384 floats (64 KB)
//   T1: same, double buffer              (64 KB)
//   S : staged x chunk [32 rows][128]    =  4096 floats (16 KB)
#define TBUF_FLOATS 16384
#define S_OFF       32768
#define LDS_FLOATS  (S_OFF + CCOLS * D_FEAT)   // 36864 floats = 144 KB

#define AS1 __attribute__((address_space(1)))
#define AS3 __attribute__((address_space(3)))

typedef int v4i __attribute__((__vector_size__(16)));

#if __has_builtin(__builtin_amdgcn_global_load_async_to_lds_b128)
#define HAVE_ASYNC_LDS 1
#else
#define HAVE_ASYNC_LDS 0
#endif

static __device__ __forceinline__ unsigned ballot32(bool p) {
#if __has_builtin(__builtin_amdgcn_ballot_w32)
  return __builtin_amdgcn_ballot_w32(p);
#else
  return (unsigned)__ballot(p);
#endif
}

static __device__ __forceinline__ float4 fmax4(float4 a, float4 b) {
  return make_float4(fmaxf(a.x, b.x), fmaxf(a.y, b.y),
                     fmaxf(a.z, b.z), fmaxf(a.w, b.w));
}

static __device__ __forceinline__ void async_wait_all() {
#if HAVE_ASYNC_LDS
  asm volatile("s_wait_asynccnt 0x0" ::: "memory");
#endif
}

__global__ __launch_bounds__(BLOCK) void MaxPoolAgg_kernel(
    const float* __restrict__ x,
    const int*   __restrict__ adj,
    float*       __restrict__ out) {
  extern __shared__ __align__(16) float lds[];
  float* const S = lds + S_OFF;

  const int tid  = threadIdx.x;
  const int wave = tid >> 5;
  const int lane = tid & 31;
  const int row0 = blockIdx.x * ROWS_PER_WG + wave * ROWS_PER_WAVE;

  const float NEG = -3.402823466e+38f;  // jnp.finfo(f32).min
  const float4 NEG4 = make_float4(NEG, NEG, NEG, NEG);

  float4   acc[ROWS_PER_WAVE];
  unsigned any[ROWS_PER_WAVE];
#pragma unroll
  for (int r = 0; r < ROWS_PER_WAVE; ++r) { acc[r] = NEG4; any[r] = 0u; }

  // ---- stage(c): async-copy x[c*CCOLS ..][:] (16 KB) into S ----
  auto stage = [&](int c) {
    const float* gsrc = x + (size_t)c * CCOLS * D_FEAT;
#pragma unroll
    for (int k = 0; k < (CCOLS * D_FEAT) / (BLOCK * 4); ++k) {  // 4 x 16B / thread
      const int u = tid + k * BLOCK;  // float4 unit
#if HAVE_ASYNC_LDS
      __builtin_amdgcn_global_load_async_to_lds_b128(
          (AS1 v4i*)(gsrc + (size_t)u * 4),
          (AS3 v4i*)(S + u * 4), 0, 0);
#else
      *(float4*)(S + u * 4) = *(const float4*)(gsrc + (size_t)u * 4);
#endif
    }
  };

  // ---- build: subset-max table for the chunk currently in S ----
  // Wave w builds group w (columns 4w..4w+3): T[g][m] = max over set bits k of x col (4g+k).
  auto build = [&](float* T) {
    const int g = wave;  // 8 waves == 8 groups
    float4 C[4];
#pragma unroll
    for (int k = 0; k < 4; ++k)
      C[k] = *(const float4*)(S + (g * 4 + k) * D_FEAT + lane * 4);
    float4 E[16];
    E[0] = NEG4;
#pragma unroll
    for (int m = 1; m < 16; ++m)
      E[m] = fmax4(E[m & (m - 1)], C[__builtin_ctz((unsigned)m)]);
#pragma unroll
    for (int m = 0; m < 16; ++m)
      *(float4*)(T + (g * 16 + m) * D_FEAT + lane * 4) = E[m];
  };

  // ---- prologue: fill S with chunk 0, build T[0] ----
  stage(0);
  async_wait_all();
  __syncthreads();
  build(lds);  // T buffer 0 is at offset 0
  __syncthreads();

  for (int c = 0; c < NCHUNK; ++c) {
    const float* T = lds + (c & 1) * TBUF_FLOATS;
    if (c + 1 < NCHUNK) stage(c + 1);   // async; overlaps consume below

    // ---- consume chunk c: branch-free table-driven masked max ----
    const int cbase = c * CCOLS;
#pragma unroll
    for (int r = 0; r < ROWS_PER_WAVE; ++r) {
      // Non-temporal: 256MB adj stream must not evict x / tables from caches.
      int a = __builtin_nontemporal_load(adj + (size_t)(row0 + r) * N_NODES + cbase + lane);
      unsigned m = ballot32(a != 0);   // bit L <-> column cbase+L (uniform SGPR)
      any[r] |= m;
#pragma unroll
      for (int g = 0; g < GROUPS; g += 2) {
        unsigned n0 = (m >> (4 * g)) & 15u;
        unsigned n1 = (m >> (4 * g + 4)) & 15u;
        float4 v0 = *(const float4*)(T + (g * 16 + (int)n0) * D_FEAT + lane * 4);
        float4 v1 = *(const float4*)(T + ((g + 1) * 16 + (int)n1) * D_FEAT + lane * 4);
        // 3-operand shape -> v_max3_f32
        acc[r].x = fmaxf(fmaxf(v0.x, v1.x), acc[r].x);
        acc[r].y = fmaxf(fmaxf(v0.y, v1.y), acc[r].y);
        acc[r].z = fmaxf(fmaxf(v0.z, v1.z), acc[r].z);
        acc[r].w = fmaxf(fmaxf(v0.w, v1.w), acc[r].w);
      }
    }

    async_wait_all();       // S now holds chunk c+1
    __syncthreads();        // everyone done with T[1-p] (reads) and S (writes)
    if (c + 1 < NCHUNK) build(lds + ((c + 1) & 1) * TBUF_FLOATS);
    __syncthreads();        // table for c+1 visible before it is consumed
  }

#pragma unroll
  for (int r = 0; r < ROWS_PER_WAVE; ++r) {
    float4 res = acc[r];
    if (any[r] == 0u) res = make_float4(0.f, 0.f, 0.f, 0.f);  // no neighbors -> zeros
    *(float4*)(out + (size_t)(row0 + r) * D_FEAT + lane * 4) = res;
  }
}

extern "C" void kernel_launch(void* const* d_in, const int* in_sizes, int n_in,
                              void* d_out, int out_size, void* d_ws, size_t ws_size,
                              hipStream_t stream) {
  const float* x   = (const float*)d_in[0];   // [8192, 128] f32
  const int*   adj = (const int*)d_in[1];     // [8192, 8192] i32
  float*       out = (float*)d_out;           // [8192, 128] f32
  (void)in_sizes; (void)n_in; (void)out_size; (void)d_ws; (void)ws_size;

  MaxPoolAgg_kernel<<<dim3(GRID), dim3(BLOCK), (size_t)LDS_FLOATS * 4, stream>>>(x, adj, out);
}